// TangentVectorizer_5746666242189
// MI455X (gfx1250) — compile-verified
//
#include <hip/hip_runtime.h>
#include <hip/hip_bf16.h>
#include <math.h>

typedef float v2f __attribute__((ext_vector_type(2)));
typedef float v8f __attribute__((ext_vector_type(8)));
typedef unsigned int v4u __attribute__((ext_vector_type(4)));
typedef int v4i __attribute__((ext_vector_type(4)));
typedef int v8i __attribute__((ext_vector_type(8)));

#define LDX 68   // padded LDS row stride (floats): 16B-aligned rows, spreads banks

#if defined(__has_builtin)
#if __has_builtin(__builtin_amdgcn_tensor_load_to_lds) && __has_builtin(__builtin_amdgcn_s_wait_tensorcnt)
#define HAVE_TDM 1
#endif
#endif
#ifndef HAVE_TDM
#define HAVE_TDM 0
#endif

// map linear upper-triangular index (row-major, r<=c, n=64) -> (r,c)
__device__ __forceinline__ void tri_rc(int idx, int& r, int& c) {
  float fr = (129.0f - sqrtf(16641.0f - 8.0f * (float)idx)) * 0.5f;
  r = (int)fr;
  if (r < 0) r = 0;
  if (r > 63) r = 63;
  while (r > 0 && r * (129 - r) / 2 > idx) --r;
  while (r < 63 && (r + 1) * (129 - (r + 1)) / 2 <= idx) ++r;
  c = r + (idx - r * (129 - r) / 2);
}

#if HAVE_TDM
// TDM: load a [128 x 64] fp32 tile (row stride 4096 elems) into LDS with
// per-row padding of 4 DWORDs -> LDS row stride 68 floats.
__device__ __forceinline__ void tdm_load_xtile(const float* gsrc, unsigned int lds_byte_off) {
  unsigned long long ga = (unsigned long long)(uintptr_t)gsrc;
  v4u g0;
  g0.x = 1u;                                              // count=1 (valid user D#)
  g0.y = lds_byte_off;                                    // lds_addr (bytes)
  g0.z = (unsigned int)(ga & 0xFFFFFFFFu);                // global_addr[31:0]
  g0.w = (unsigned int)((ga >> 32) & 0x01FFFFFFu)         // global_addr[56:32]
       | (2u << 30);                                      // type = 2 ("image")
  v8i g1;
  g1[0] = (int)((2u << 16)                                // data_size = 4 bytes
              | (1u << 20)                                // pad_enable
              | (5u << 22)                                // pad_interval: 64 DWORDs
              | (3u << 25));                              // pad_amount: 4 DWORDs
  g1[1] = (int)(4096u << 16);                             // tensor_dim0[15:0] = 4096
  g1[2] = (int)(128u << 16);                              // dim0[31:16]=0 | tensor_dim1 = 128
  g1[3] = (int)(64u << 16);                               // dim1[31:16]=0 | tile_dim0 = 64
  g1[4] = 128;                                            // tile_dim1 = 128 | tile_dim2 = 0
  g1[5] = 4096;                                           // tensor_dim0_stride[31:0]
  g1[6] = 0;                                              // stride[47:32] | dim1_stride[15:0]
  g1[7] = 0;
  v4i z4 = {0, 0, 0, 0};
#if __clang_major__ >= 23
  v8i z8 = {0, 0, 0, 0, 0, 0, 0, 0};
  __builtin_amdgcn_tensor_load_to_lds(g0, g1, z4, z4, z8, 0);
#else
  __builtin_amdgcn_tensor_load_to_lds(g0, g1, z4, z4, 0);
#endif
}
#endif

__global__ __launch_bounds__(256) void tangent_vec_kernel(
    const float* __restrict__ x,      // [B,128,1,4096]
    const float* __restrict__ w,      // [64,128]
    float* __restrict__ out)          // [B,2080]
{
  constexpr int C = 128, F = 64, T = 4096, TT = 64;
  constexpr int NSWEEP = 8;

  __shared__ union {
    struct { float xs[C][LDX]; float ys[F][LDX]; } g;                  // GEMM phase
    struct { float aM[F][LDX]; float vM[F][LDX];                      // Jacobi phase
             float cs[32]; float sn[32]; int pp[32]; int qq[32];
             float le[F]; } j;
  } sm;
  __shared__ float sy[F];
  __shared__ float muS;

  const int t    = threadIdx.x;
  const int b    = blockIdx.x;
  const int lane = t & 31;
  const int wv   = t >> 5;       // wave 0..7
  const int lo   = lane & 15;
  const int hi   = lane >> 4;

  // ---- GEMM1 (Y = W*X): wave wv -> output rows [M0,M0+16), col tiles N0a,N0b
  const int M0  = (wv >> 1) * 16;
  const int N0a = (wv & 1) * 32;
  const int N0b = N0a + 16;

  // Cache A-fragments of W for this wave's row block: K=128 -> 32 fragments.
  // A 16x4 fp32 layout: lanes 0-15 row M=lo hold K=k0,k0+1; lanes 16-31 hold K=k0+2,k0+3.
  v2f aW[32];
  #pragma unroll
  for (int kk = 0; kk < 32; ++kk) {
    const float* wp = w + (M0 + lo) * C + kk * 4 + 2 * hi;
    aW[kk].x = wp[0];
    aW[kk].y = wp[1];
  }

  // ---- GEMM2 (cov += Y*Y^T): wave owns cov tiles (Ma0,Nc0) and (Mb0,Nc0) (shared B frag)
  const int Ma0 = (wv >> 2) * 16;
  const int Mb0 = Ma0 + 32;
  const int Nc0 = (wv & 3) * 16;

  v8f acc0 = {0.f,0.f,0.f,0.f,0.f,0.f,0.f,0.f};
  v8f acc1 = acc0;

  if (t < F) sy[t] = 0.f;
  if (t == 0) muS = 0.f;

  const float* xb = x + (size_t)b * C * T;

#if HAVE_TDM
  const unsigned int xs_off = (unsigned int)(uintptr_t)(&sm.g.xs[0][0]);
  if (wv == 0) tdm_load_xtile(xb, xs_off);                 // prefetch tile 0 via TDM
#else
  // fallback: stage tile 0 with vector loads
  #pragma unroll
  for (int i = 0; i < 8; ++i) {
    int idv = i * 256 + t;
    int cc  = idv >> 4;
    int jj  = (idv & 15) << 2;
    const float4 v = *(const float4*)(xb + (size_t)cc * T + jj);
    sm.g.xs[cc][jj + 0] = v.x;
    sm.g.xs[cc][jj + 1] = v.y;
    sm.g.xs[cc][jj + 2] = v.z;
    sm.g.xs[cc][jj + 3] = v.w;
  }
#endif

  // ================= streaming GEMM phase (software-pipelined) =================
  for (int t0 = 0; t0 < T; t0 += TT) {
#if HAVE_TDM
    if (wv == 0) __builtin_amdgcn_s_wait_tensorcnt(0);     // tile t0 resident in xs
#endif
    __syncthreads();

    // GEMM1: Y = W * X  (M=64,N=64,K=128), 2 tiles per wave
    v8f c0 = {0.f,0.f,0.f,0.f,0.f,0.f,0.f,0.f};
    v8f c1 = c0;
    #pragma unroll
    for (int kk = 0; kk < 32; ++kk) {
      int k0 = kk * 4;
      v2f bA, bB;
      bA.x = sm.g.xs[k0 + 2*hi + 0][N0a + lo];
      bA.y = sm.g.xs[k0 + 2*hi + 1][N0a + lo];
      bB.x = sm.g.xs[k0 + 2*hi + 0][N0b + lo];
      bB.y = sm.g.xs[k0 + 2*hi + 1][N0b + lo];
      c0 = __builtin_amdgcn_wmma_f32_16x16x4_f32(false, aW[kk], false, bA, (short)0, c0, false, false);
      c1 = __builtin_amdgcn_wmma_f32_16x16x4_f32(false, aW[kk], false, bB, (short)0, c1, false, false);
    }
    // store Y tile (C/D layout: VGPR v -> row v+8*hi, col lo)
    #pragma unroll
    for (int v = 0; v < 8; ++v) {
      sm.g.ys[M0 + v + 8*hi][N0a + lo] = c0[v];
      sm.g.ys[M0 + v + 8*hi][N0b + lo] = c1[v];
    }
    __syncthreads();   // all waves done reading xs + ys tile visible

    // ---- xs is now dead: launch next tile's copy, overlapped with GEMM2 ----
    if (t0 + TT < T) {
#if HAVE_TDM
      if (wv == 0) tdm_load_xtile(xb + t0 + TT, xs_off);
#else
      #pragma unroll
      for (int i = 0; i < 8; ++i) {
        int idv = i * 256 + t;
        int cc  = idv >> 4;
        int jj  = (idv & 15) << 2;
        const float4 v = *(const float4*)(xb + (size_t)cc * T + t0 + TT + jj);
        sm.g.xs[cc][jj + 0] = v.x;
        sm.g.xs[cc][jj + 1] = v.y;
        sm.g.xs[cc][jj + 2] = v.z;
        sm.g.xs[cc][jj + 3] = v.w;
      }
#endif
    }

    // row sums of Y tile (for mean centering identity)
    {
      int r = t & 63, qd = t >> 6;
      float s = 0.f;
      #pragma unroll
      for (int jj = 0; jj < 16; ++jj) s += sm.g.ys[r][qd * 16 + jj];
      atomicAdd(&sy[r], s);
    }
    // GEMM2: cov += Y * Y^T  (M=64,N=64,K=64); B fragment shared by both tiles
    #pragma unroll
    for (int kk = 0; kk < 16; ++kk) {
      int k0 = kk * 4;
      v2f aA, aB, bC;
      aA.x = sm.g.ys[Ma0 + lo][k0 + 2*hi];  aA.y = sm.g.ys[Ma0 + lo][k0 + 2*hi + 1];
      aB.x = sm.g.ys[Mb0 + lo][k0 + 2*hi];  aB.y = sm.g.ys[Mb0 + lo][k0 + 2*hi + 1];
      bC.x = sm.g.ys[Nc0 + lo][k0 + 2*hi];  bC.y = sm.g.ys[Nc0 + lo][k0 + 2*hi + 1];
      acc0 = __builtin_amdgcn_wmma_f32_16x16x4_f32(false, aA, false, bC, (short)0, acc0, false, false);
      acc1 = __builtin_amdgcn_wmma_f32_16x16x4_f32(false, aB, false, bC, (short)0, acc1, false, false);
    }
    // no trailing barrier: loop-top wait+barrier retires the async tile copy
  }
  __syncthreads();

  // ================= finalize covariance in LDS =================
  #pragma unroll
  for (int v = 0; v < 8; ++v) {
    sm.j.aM[Ma0 + v + 8*hi][Nc0 + lo] = acc0[v];
    sm.j.aM[Mb0 + v + 8*hi][Nc0 + lo] = acc1[v];
  }
  __syncthreads();

  const float invT   = 1.0f / (float)T;
  const float invTm1 = 1.0f / (float)(T - 1);
  #pragma unroll
  for (int i = 0; i < 16; ++i) {
    int idx = i * 256 + t;
    int r = idx >> 6, cc = idx & 63;
    sm.j.aM[r][cc] = (sm.j.aM[r][cc] - sy[r] * sy[cc] * invT) * invTm1;
    sm.j.vM[r][cc] = (r == cc) ? 1.f : 0.f;
  }
  __syncthreads();
  if (t < 64) atomicAdd(&muS, sm.j.aM[t][t]);
  __syncthreads();
  const float mu = muS * (1.0f / 64.0f);

  // shrinkage + symmetrize + jitter (each thread owns upper-tri pairs)
  for (int idx = t; idx < 2080; idx += 256) {
    int r, cc; tri_rc(idx, r, cc);
    if (r == cc) {
      sm.j.aM[r][r] = 0.9f * sm.j.aM[r][r] + 0.1f * mu + 1e-4f;
    } else {
      float v0 = 0.45f * (sm.j.aM[r][cc] + sm.j.aM[cc][r]);   // 0.5*(1-alpha)
      sm.j.aM[r][cc] = v0;
      sm.j.aM[cc][r] = v0;
    }
  }
  __syncthreads();

  // ================= cyclic Jacobi eigensolver =================
  for (int sweep = 0; sweep < NSWEEP; ++sweep) {
    for (int r = 0; r < 63; ++r) {
      if (t < 32) {   // one thread per rotation pair (circle method)
        int p, q;
        if (t == 0) { p = 63; q = r; }
        else        { p = (r + t) % 63; q = (r - t + 63) % 63; }
        sm.j.pp[t] = p; sm.j.qq[t] = q;
        float app = sm.j.aM[p][p], aqq = sm.j.aM[q][q], apq = sm.j.aM[p][q];
        float c = 1.f, s = 0.f;
        if (fabsf(apq) > 1e-20f) {
          float tau = (aqq - app) / (2.0f * apq);
          float tt  = (tau >= 0.f) ? 1.0f / (tau + sqrtf(1.0f + tau * tau))
                                   : 1.0f / (tau - sqrtf(1.0f + tau * tau));
          c = 1.0f / sqrtf(1.0f + tt * tt);
          s = tt * c;
        }
        sm.j.cs[t] = c; sm.j.sn[t] = s;
      }
      __syncthreads();
      {  // row phase: A <- J^T A   (8 threads per pair, 8 cols each)
        int m = t >> 3;
        int p = sm.j.pp[m], q = sm.j.qq[m];
        float c = sm.j.cs[m], s = sm.j.sn[m];
        #pragma unroll
        for (int u = 0; u < 8; ++u) {
          int jj = (t & 7) + u * 8;
          float ap = sm.j.aM[p][jj], aq = sm.j.aM[q][jj];
          sm.j.aM[p][jj] = c * ap - s * aq;
          sm.j.aM[q][jj] = s * ap + c * aq;
        }
      }
      __syncthreads();
      {  // column phase: A <- A J, V <- V J
        int m = t >> 3;
        int p = sm.j.pp[m], q = sm.j.qq[m];
        float c = sm.j.cs[m], s = sm.j.sn[m];
        #pragma unroll
        for (int u = 0; u < 8; ++u) {
          int ii = (t & 7) + u * 8;
          float ap = sm.j.aM[ii][p], aq = sm.j.aM[ii][q];
          sm.j.aM[ii][p] = c * ap - s * aq;
          sm.j.aM[ii][q] = s * ap + c * aq;
          float vp = sm.j.vM[ii][p], vq = sm.j.vM[ii][q];
          sm.j.vM[ii][p] = c * vp - s * vq;
          sm.j.vM[ii][q] = s * vp + c * vq;
        }
      }
      __syncthreads();
    }
  }

  // ================= log map + upper-tri vectorization =================
  if (t < 64) sm.j.le[t] = logf(fmaxf(sm.j.aM[t][t], 1e-4f));
  __syncthreads();

  float* ob = out + (size_t)b * 2080;
  for (int idx = t; idx < 2080; idx += 256) {
    int r, cc; tri_rc(idx, r, cc);
    float s = 0.f;
    #pragma unroll 8
    for (int k = 0; k < 64; ++k)
      s += sm.j.vM[r][k] * sm.j.le[k] * sm.j.vM[cc][k];
    ob[idx] = (r == cc) ? s : s * 1.41421356237309515f;
  }
}

extern "C" void kernel_launch(void* const* d_in, const int* in_sizes, int n_in,
                              void* d_out, int out_size, void* d_ws, size_t ws_size,
                              hipStream_t stream) {
  const float* x = (const float*)d_in[0];     // [B,128,1,4096] fp32
  const float* w = (const float*)d_in[1];     // [64,128] fp32
  float* out     = (float*)d_out;             // [B,2080] fp32
  int B = in_sizes[0] / (128 * 4096);
  tangent_vec_kernel<<<B, 256, 0, stream>>>(x, w, out);
}